// Model_85624468013125
// MI455X (gfx1250) — compile-verified
//
#include <hip/hip_runtime.h>

typedef float v2f __attribute__((ext_vector_type(2)));
typedef float v8f __attribute__((ext_vector_type(8)));

#define BN_EPS 1e-5f

__device__ __forceinline__ v8f wmma4(v2f a, v2f b, v8f c) {
  // D = A(16x4 f32) * B(4x16 f32) + C(16x16 f32)
  return __builtin_amdgcn_wmma_f32_16x16x4_f32(false, a, false, b, (short)0, c, false, false);
}

// ---------------------------------------------------------------------------
// 1) Projection GEMM: pro[b] = relu(bn(W(256x2048) @ x[b](2048x784)))
//    grid(13,4,32) block(128): 4 waves, each wave -> 16(M) x 64(N) strip
// ---------------------------------------------------------------------------
__global__ void proj_gemm_kernel(const float* __restrict__ x, const float* __restrict__ W,
                                 const float* __restrict__ gamma, const float* __restrict__ beta,
                                 const float* __restrict__ bnmean, const float* __restrict__ bnvar,
                                 float* __restrict__ pro) {
  const int b    = blockIdx.z;
  const int lane = threadIdx.x & 31;
  const int wave = threadIdx.x >> 5;
  const int half = lane >> 4;
  const int r    = lane & 15;
  const int m0     = blockIdx.y * 64 + wave * 16;
  const int n0base = blockIdx.x * 64;
  const float* __restrict__ X    = x + (size_t)b * 2048 * 784;
  const float* __restrict__ Arow = W + (size_t)(m0 + r) * 2048;

  v8f acc[4] = {};
  for (int k = 0; k < 2048; k += 4) {
    const int ka = k + half * 2;
    if ((k & 63) == 0 && k + 64 < 2048) {
      __builtin_prefetch(Arow + ka + 64, 0, 1);  // global_prefetch_b8, next W chunk
    }
    v2f af = { Arow[ka], Arow[ka + 1] };
#pragma unroll
    for (int t = 0; t < 4; ++t) {
      int n = n0base + t * 16 + r;
      n = n < 784 ? n : 783;  // clamp loads on edge tile; stores guarded below
      v2f bf = { X[(size_t)ka * 784 + n], X[(size_t)(ka + 1) * 784 + n] };
      acc[t] = wmma4(af, bf, acc[t]);
    }
  }
  // BN + ReLU epilogue
  float iv[8], bs[8];
#pragma unroll
  for (int j = 0; j < 8; ++j) {
    const int row = m0 + half * 8 + j;
    iv[j] = gamma[row] * rsqrtf(bnvar[row] + BN_EPS);
    bs[j] = beta[row] - bnmean[row] * iv[j];
  }
#pragma unroll
  for (int t = 0; t < 4; ++t) {
    const int col = n0base + t * 16 + r;
    if (col < 784) {
#pragma unroll
      for (int j = 0; j < 8; ++j) {
        const int row = m0 + half * 8 + j;
        float v = fmaxf(acc[t][j] * iv[j] + bs[j], 0.0f);
        pro[((size_t)b * 256 + row) * 784 + col] = v;
      }
    }
  }
}

// ---------------------------------------------------------------------------
// 2) Channel means g[b][c] = mean_hw(pro) (feeds both attention and cov centering)
// ---------------------------------------------------------------------------
__global__ void chan_mean_kernel(const float* __restrict__ pro, float* __restrict__ g) {
  const int b = blockIdx.y, c = blockIdx.x;
  const float* __restrict__ p = pro + ((size_t)b * 256 + c) * 784;
  float s = 0.0f;
  for (int i = threadIdx.x; i < 784; i += 128) s += p[i];
  __shared__ float sh[128];
  sh[threadIdx.x] = s;
  __syncthreads();
  for (int off = 64; off > 0; off >>= 1) {
    if ((int)threadIdx.x < off) sh[threadIdx.x] += sh[threadIdx.x + off];
    __syncthreads();
  }
  if (threadIdx.x == 0) g[b * 256 + c] = sh[0] * (1.0f / 784.0f);
}

// ---------------------------------------------------------------------------
// 3) Attention: att = sigmoid(W2 @ (W1 @ g))
// ---------------------------------------------------------------------------
__global__ void att_kernel(const float* __restrict__ g, const float* __restrict__ w1,
                           const float* __restrict__ w2, float* __restrict__ att) {
  const int b = blockIdx.x, t = threadIdx.x;
  __shared__ float sg[256], sa1[16];
  sg[t] = g[b * 256 + t];
  __syncthreads();
  if (t < 16) {
    float s = 0.0f;
    for (int c = 0; c < 256; ++c) s += w1[t * 256 + c] * sg[c];
    sa1[t] = s;
  }
  __syncthreads();
  float s = 0.0f;
#pragma unroll
  for (int j = 0; j < 16; ++j) s += w2[t * 16 + j] * sa1[j];
  att[b * 256 + t] = 1.0f / (1.0f + __expf(-s));
}

__global__ void satt_kernel(const float* __restrict__ att, float* __restrict__ satt) {
  const int b = blockIdx.y, i = blockIdx.x, j = threadIdx.x;
  satt[((size_t)b * 256 + i) * 256 + j] = att[b * 256 + i] * att[b * 256 + j];
}

// ---------------------------------------------------------------------------
// 4) Covariance: A[b] = (pro[b]-mean)(pro[b]-mean)^T / 784 -- grid(4,4,32) blk 128
//    B operand is pro^T -> contiguous float2 per lane (K contiguous)
// ---------------------------------------------------------------------------
__global__ void cov_gemm_kernel(const float* __restrict__ pro, const float* __restrict__ g,
                                float* __restrict__ Aout) {
  const int b    = blockIdx.z;
  const int lane = threadIdx.x & 31;
  const int wave = threadIdx.x >> 5;
  const int half = lane >> 4;
  const int r    = lane & 15;
  const int m0     = blockIdx.y * 64 + wave * 16;
  const int n0base = blockIdx.x * 64;
  const float* __restrict__ P  = pro + (size_t)b * 256 * 784;
  const float* __restrict__ gm = g + b * 256;
  const float sa = gm[m0 + r];
  float sb[4];
#pragma unroll
  for (int t = 0; t < 4; ++t) sb[t] = gm[n0base + t * 16 + r];

  v8f acc[4] = {};
  const float* __restrict__ Arow = P + (size_t)(m0 + r) * 784;
  for (int k = 0; k < 784; k += 4) {
    const int ka = k + half * 2;
    v2f af = { Arow[ka] - sa, Arow[ka + 1] - sa };
#pragma unroll
    for (int t = 0; t < 4; ++t) {
      const int n = n0base + t * 16 + r;
      v2f bf = { P[(size_t)n * 784 + ka] - sb[t], P[(size_t)n * 784 + ka + 1] - sb[t] };
      acc[t] = wmma4(af, bf, acc[t]);
    }
  }
#pragma unroll
  for (int t = 0; t < 4; ++t) {
    const int col = n0base + t * 16 + r;
#pragma unroll
    for (int j = 0; j < 8; ++j) {
      const int row = m0 + half * 8 + j;
      Aout[((size_t)b * 256 + row) * 256 + col] = acc[t][j] * (1.0f / 784.0f);
    }
  }
}

// ---------------------------------------------------------------------------
// 5) trace(A) per batch
// ---------------------------------------------------------------------------
__global__ void trace_kernel(const float* __restrict__ A, float* __restrict__ normA) {
  const int b = blockIdx.x, t = threadIdx.x;
  __shared__ float sh[256];
  sh[t] = A[(size_t)b * 65536 + t * 257];
  __syncthreads();
  for (int off = 128; off > 0; off >>= 1) {
    if (t < off) sh[t] += sh[t + off];
    __syncthreads();
  }
  if (t == 0) normA[b] = sh[0];
}

// ---------------------------------------------------------------------------
// 6) momn elementwise phases -- grid(256,32) block(256)
// ---------------------------------------------------------------------------
__global__ void momn_init_kernel(const float* __restrict__ A, const float* __restrict__ satt,
                                 const float* __restrict__ normA,
                                 float* __restrict__ J1, float* __restrict__ J2,
                                 float* __restrict__ Yp, float* __restrict__ T1,
                                 float* __restrict__ Z,
                                 float e1b1, float e2b2, float m1e3, float m2e3) {
  const int b = blockIdx.y, row = blockIdx.x, col = threadIdx.x;
  const size_t idx = ((size_t)b * 256 + row) * 256 + col;
  const bool diag = (row == col);
  const float an = A[idx] / normA[b];
  const float j1 = diag ? an * (1.0f - e1b1) : an;
  const float j2 = an * (1.0f - e2b2 * (1.0f - satt[idx]));
  const float yp = an + m1e3 * (j1 - an) + m2e3 * (j2 - an);
  const float zy = 0.5f * ((diag ? 3.0f : 0.0f) - an);
  J1[idx] = j1; J2[idx] = j2; Yp[idx] = yp; T1[idx] = zy; Z[idx] = zy;
}

__global__ void momn_emid_kernel(float* __restrict__ J1, float* __restrict__ J2,
                                 const float* __restrict__ Y,
                                 const float* __restrict__ L1, const float* __restrict__ L2,
                                 const float* __restrict__ satt, float* __restrict__ Yp,
                                 float e1m1, float eta1, float b1e1,
                                 float e2m2, float eta2, float b2e2,
                                 float m1e3, float m2e3, float eta3) {
  const int b = blockIdx.y, row = blockIdx.x, col = threadIdx.x;
  const size_t idx = ((size_t)b * 256 + row) * 256 + col;
  const bool diag = (row == col);
  const float y = Y[idx], l1 = L1[idx], l2 = L2[idx];
  float j1 = J1[idx];
  j1 = j1 - e1m1 * (j1 - y) - eta1 * l1;
  if (diag) j1 *= (1.0f - b1e1);
  float j2 = J2[idx];
  j2 = j2 - e2m2 * (j2 - y) - eta2 * l2;
  j2 *= (1.0f - b2e2 * (1.0f - satt[idx]));
  const float yp = y + m1e3 * (j1 - y) + m2e3 * (j2 - y) + eta3 * (l1 + l2);
  J1[idx] = j1; J2[idx] = j2; Yp[idx] = yp;
}

template <int ACC>
__global__ void lupd_kernel(float* __restrict__ L1, float* __restrict__ L2,
                            const float* __restrict__ J1, const float* __restrict__ J2,
                            const float* __restrict__ Y, float mu1, float mu2) {
  const size_t idx = ((size_t)blockIdx.y * 256 + blockIdx.x) * 256 + threadIdx.x;
  const float y = Y[idx];
  const float a1 = mu1 * (J1[idx] - y);
  const float a2 = mu2 * (J2[idx] - y);
  L1[idx] = ACC ? (L1[idx] + a1) : a1;
  L2[idx] = ACC ? (L2[idx] + a2) : a2;
}

// ---------------------------------------------------------------------------
// 7) batched 256x256x256 momn GEMM -- grid(4,4,32) block(128)
//    MODE 0: C = P@Q ; MODE 1: C = 0.5*(3I - P@Q)
//    B k-panels (16x64) are staged to LDS once per block through the CDNA5
//    async global->LDS path (all 4 waves share the same B fragments; direct
//    loads would fetch them 4x). ASYNCcnt fenced with s_wait_asynccnt.
// ---------------------------------------------------------------------------
template <int MODE>
__global__ void momn_gemm_kernel(const float* __restrict__ P, const float* __restrict__ Q,
                                 float* __restrict__ C) {
  __shared__ float Bsh[16 * 64];  // [krel][n] panel, 4 KB
  const int b    = blockIdx.z;
  const int tid  = threadIdx.x;
  const int lane = tid & 31;
  const int wave = tid >> 5;
  const int half = lane >> 4;
  const int r    = lane & 15;
  const int m0     = blockIdx.y * 64 + wave * 16;
  const int n0base = blockIdx.x * 64;
  const float* __restrict__ A = P + (size_t)b * 65536;
  float* __restrict__ Cb = C + (size_t)b * 65536;
  // panel base for async staging: column window [n0base, n0base+64)
  const float* Bpanel = Q + (size_t)b * 65536 + n0base;

  // staging geometry: 128 threads x b128 -> 8 rows of 64 floats per issue
  const int srow  = tid >> 4;          // 0..7
  const int scol4 = (tid & 15) * 4;    // 0,4,...,60
  const unsigned lds0 = (unsigned)(unsigned long long)(const void*)&Bsh[srow * 64 + scol4];
  const unsigned lds1 = (unsigned)(unsigned long long)(const void*)&Bsh[(srow + 8) * 64 + scol4];

  v8f acc[4] = {};
  const float* __restrict__ Arow = A + (size_t)(m0 + r) * 256;

  for (int kb = 0; kb < 256; kb += 16) {
    const unsigned g0 = (unsigned)(((kb + srow) * 256 + scol4) * 4);        // byte offsets
    const unsigned g1 = (unsigned)(((kb + srow + 8) * 256 + scol4) * 4);
    asm volatile("global_load_async_to_lds_b128 %0, %1, %2"
                 :: "v"(lds0), "v"(g0), "s"(Bpanel) : "memory");
    asm volatile("global_load_async_to_lds_b128 %0, %1, %2"
                 :: "v"(lds1), "v"(g1), "s"(Bpanel) : "memory");
    asm volatile("s_wait_asynccnt 0" ::: "memory");
    __syncthreads();
#pragma unroll
    for (int k = 0; k < 16; k += 4) {
      const int ka = k + half * 2;
      v2f af = { Arow[kb + ka], Arow[kb + ka + 1] };
#pragma unroll
      for (int t = 0; t < 4; ++t) {
        const int n = t * 16 + r;
        v2f bf = { Bsh[ka * 64 + n], Bsh[(ka + 1) * 64 + n] };
        acc[t] = wmma4(af, bf, acc[t]);
      }
    }
    __syncthreads();
  }

#pragma unroll
  for (int t = 0; t < 4; ++t) {
    const int col = n0base + t * 16 + r;
#pragma unroll
    for (int j = 0; j < 8; ++j) {
      const int row = m0 + half * 8 + j;
      float v = acc[t][j];
      if (MODE == 1) v = (row == col ? 1.5f : 0.0f) - 0.5f * v;
      Cb[(size_t)row * 256 + col] = v;
    }
  }
}

// ---------------------------------------------------------------------------
// 8) tril gather: feat[b][r*(r+1)/2 + c] = Y[b][r][c]*sqrt(normA[b]), c<=r
// ---------------------------------------------------------------------------
__global__ void feat_kernel(const float* __restrict__ Y, const float* __restrict__ normA,
                            float* __restrict__ feat) {
  const int b = blockIdx.y, r = blockIdx.x, c = threadIdx.x;
  if (c <= r) {
    const float s = sqrtf(normA[b]);
    feat[(size_t)b * 32896 + (size_t)r * (r + 1) / 2 + c] =
        Y[((size_t)b * 256 + r) * 256 + c] * s;
  }
}

// ---------------------------------------------------------------------------
// 9) FC: logit(32x200) = feat(32x32896) @ fc_w^T + b -- grid(13,2) block(32)
// ---------------------------------------------------------------------------
__global__ void fc_gemm_kernel(const float* __restrict__ feat, const float* __restrict__ fcw,
                               const float* __restrict__ fcb, float* __restrict__ logit) {
  const int lane = threadIdx.x & 31;
  const int half = lane >> 4;
  const int r    = lane & 15;
  const int m0 = blockIdx.y * 16;
  const int n0 = blockIdx.x * 16;
  int n = n0 + r; n = n < 200 ? n : 199;  // clamp loads; stores guarded
  const float* __restrict__ Arow = feat + (size_t)(m0 + r) * 32896;
  const float* __restrict__ Brow = fcw + (size_t)n * 32896;  // fc_w^T access: K contiguous

  v8f acc = {};
  for (int k = 0; k < 32896; k += 4) {
    const int ka = k + half * 2;
    v2f af = { Arow[ka], Arow[ka + 1] };
    v2f bf = { Brow[ka], Brow[ka + 1] };
    acc = wmma4(af, bf, acc);
  }
  const int col = n0 + r;
  if (col < 200) {
#pragma unroll
    for (int j = 0; j < 8; ++j) {
      const int row = m0 + half * 8 + j;  // row == batch index (0..31)
      logit[row * 200 + col] = acc[j] + fcb[col];
    }
  }
}

// ---------------------------------------------------------------------------
extern "C" void kernel_launch(void* const* d_in, const int* in_sizes, int n_in,
                              void* d_out, int out_size, void* d_ws, size_t ws_size,
                              hipStream_t stream) {
  const float* x     = (const float*)d_in[0];
  const float* projw = (const float*)d_in[1];
  const float* gamma = (const float*)d_in[2];
  const float* beta  = (const float*)d_in[3];
  const float* bnmean= (const float*)d_in[4];
  const float* bnvar = (const float*)d_in[5];
  const float* attw1 = (const float*)d_in[6];
  const float* attw2 = (const float*)d_in[7];
  const float* fcw   = (const float*)d_in[8];
  const float* fcb   = (const float*)d_in[9];

  float* out   = (float*)d_out;
  float* logit = out;                 // 32*200
  float* Aout  = out + 6400;          // 32*256*256
  float* satt  = out + 2103552;       // 32*256*256
  float* feat  = out + 4200704;       // 32*32896

  float* ws    = (float*)d_ws;
  float* pro   = ws;                  // 32*256*784 = 6,422,528
  float* g     = ws + 6422528;        // 32*256
  float* attb  = g + 8192;            // 32*256
  float* normA = attb + 8192;         // 32
  const size_t MS = (size_t)32 * 65536;
  float* J1 = normA + 32;
  float* J2 = J1 + MS;
  float* L1 = J2 + MS;
  float* L2 = L1 + MS;
  float* Yp = L2 + MS;
  float* Y  = Yp + MS;
  float* Zb = Y  + MS;
  float* T1 = Zb + MS;
  float* T2 = T1 + MS;

  proj_gemm_kernel<<<dim3(13, 4, 32), 128, 0, stream>>>(x, projw, gamma, beta, bnmean, bnvar, pro);
  chan_mean_kernel<<<dim3(256, 32), 128, 0, stream>>>(pro, g);
  att_kernel<<<32, 256, 0, stream>>>(g, attw1, attw2, attb);
  satt_kernel<<<dim3(256, 32), 256, 0, stream>>>(attb, satt);
  cov_gemm_kernel<<<dim3(4, 4, 32), 128, 0, stream>>>(pro, g, Aout);
  trace_kernel<<<32, 256, 0, stream>>>(Aout, normA);

  const float beta1 = 0.5f, beta2 = 0.5f, roph = 1.1f;
  float mu1 = 1.0f, mu2 = 1.0f;
  {
    const float eta1 = 1.0f / mu1, eta2 = 1.0f / mu2, eta3 = 1.0f / (mu1 + mu2);
    momn_init_kernel<<<dim3(256, 32), 256, 0, stream>>>(
        Aout, satt, normA, J1, J2, Yp, T1, Zb,
        eta1 * beta1, eta2 * beta2, mu1 * eta3, mu2 * eta3);
    momn_gemm_kernel<0><<<dim3(4, 4, 32), 128, 0, stream>>>(Yp, T1, Y);
    lupd_kernel<0><<<dim3(256, 32), 256, 0, stream>>>(L1, L2, J1, J2, Y, mu1, mu2);
    mu1 *= roph; mu2 *= roph;
  }
  float* Zcur = Zb;
  float* Znxt = T2;
  for (int i = 1; i <= 4; ++i) {
    const float eta1 = 1.0f / mu1, eta2 = 1.0f / mu2, eta3 = 1.0f / (mu1 + mu2);
    momn_emid_kernel<<<dim3(256, 32), 256, 0, stream>>>(
        J1, J2, Y, L1, L2, satt, Yp,
        eta1 * mu1, eta1, beta1 * eta1,
        eta2 * mu2, eta2, beta2 * eta2,
        mu1 * eta3, mu2 * eta3, eta3);
    momn_gemm_kernel<1><<<dim3(4, 4, 32), 128, 0, stream>>>(Zcur, Yp, T1);  // ZY = 0.5(3I - Z@Y')
    momn_gemm_kernel<0><<<dim3(4, 4, 32), 128, 0, stream>>>(Yp, T1, Y);     // Y = Y'@ZY
    if (i < 4) {
      momn_gemm_kernel<0><<<dim3(4, 4, 32), 128, 0, stream>>>(T1, Zcur, Znxt);  // Z = ZY@Z
      float* tmp = Zcur; Zcur = Znxt; Znxt = tmp;
      lupd_kernel<1><<<dim3(256, 32), 256, 0, stream>>>(L1, L2, J1, J2, Y, mu1, mu2);
      mu1 *= roph; mu2 *= roph;
    }
  }
  feat_kernel<<<dim3(256, 32), 256, 0, stream>>>(Y, normA, feat);
  fc_gemm_kernel<<<dim3(13, 2), 32, 0, stream>>>(feat, fcw, fcb, logit);
}